// WaveFunction_56556129354666
// MI455X (gfx1250) — compile-verified
//
#include <hip/hip_runtime.h>
#include <math.h>

#define B_TOTAL 262144
#define D_IN 10
#define EPSV 1e-5f

typedef float v2f __attribute__((ext_vector_type(2)));
typedef float v4f __attribute__((ext_vector_type(4)));
typedef float v8f __attribute__((ext_vector_type(8)));

// workspace layout (float offsets)
#define OFF_W1T 0                 // 30*3*32*2 = 5760 : A-operand-layout W1all^T
#define OFF_S1  5760              // 480 BN1 scale
#define OFF_T1  6240              // 480 BN1 bias
#define OFF_S2  6720              // 480 BN2 scale
#define OFF_T2  7200              // 480 BN2 bias
#define NB1     256
#define OFF_P1  7680              // NB1*65 phase-1 partials
#define C2      64
#define OFF_P2  (OFF_P1 + NB1*65) // 30*C2*32 phase-2 partials

// ---------------------------------------------------------------------------
// Pre-pass: bake W1all^T (perm folded into W1) directly into the WMMA
// 16x4 f32 A-operand lane layout: lanes 0-15 hold K=k0,k0+1; lanes 16-31 K=k0+2,k0+3.
__global__ void k_setup_w1t(const float* __restrict__ W1, const int* __restrict__ perms,
                            float* __restrict__ ws) {
  int gid = blockIdx.x * blockDim.x + threadIdx.x;
  if (gid >= 30 * 3 * 32) return;
  int mt = gid / 96, rem = gid % 96;
  int kc = rem / 32, lane = rem % 32;
  int hi = lane >> 4, m = lane & 15;
  int row = mt * 16 + m;          // perm-channel index 0..479
  int p = row >> 2, c = row & 3;
  float vals[2];
  #pragma unroll
  for (int t = 0; t < 2; ++t) {
    int k = kc * 4 + 2 * hi + t;  // K index
    float v = 0.f;
    if (k < D_IN) {
      int q = k >> 1, d = k & 1;
      int j = 0;
      for (int jj = 0; jj < 5; ++jj) if (perms[p * 5 + jj] == q) j = jj;
      v = W1[(2 * j + d) * 4 + c];
    }
    vals[t] = v;
  }
  float* dst = ws + OFF_W1T + ((mt * 3 + kc) * 32 + lane) * 2;
  dst[0] = vals[0]; dst[1] = vals[1];
}

// ---------------------------------------------------------------------------
// Phase 1: sums and upper-tri cross products of x (10 + 55 values).
__global__ void k_phase1(const float* __restrict__ x, float* __restrict__ ws) {
  float s[10], cc[55];
  #pragma unroll
  for (int i = 0; i < 10; ++i) s[i] = 0.f;
  #pragma unroll
  for (int i = 0; i < 55; ++i) cc[i] = 0.f;
  int tid = threadIdx.x;
  for (int b = blockIdx.x * blockDim.x + tid; b < B_TOTAL; b += gridDim.x * blockDim.x) {
    float xv[10];
    #pragma unroll
    for (int k = 0; k < 10; ++k) xv[k] = x[b * 10 + k];
    #pragma unroll
    for (int k = 0; k < 10; ++k) s[k] += xv[k];
    int idx = 0;
    #pragma unroll
    for (int i = 0; i < 10; ++i)
      #pragma unroll
      for (int j = i; j < 10; ++j) { cc[idx] += xv[i] * xv[j]; ++idx; }
  }
  #pragma unroll
  for (int m = 1; m < 32; m <<= 1) {
    #pragma unroll
    for (int i = 0; i < 10; ++i) s[i] += __shfl_xor(s[i], m, 32);
    #pragma unroll
    for (int i = 0; i < 55; ++i) cc[i] += __shfl_xor(cc[i], m, 32);
  }
  __shared__ float acc[65];
  if (tid < 65) acc[tid] = 0.f;
  __syncthreads();
  if ((tid & 31) == 0) {
    for (int i = 0; i < 10; ++i) atomicAdd(&acc[i], s[i]);
    for (int i = 0; i < 55; ++i) atomicAdd(&acc[10 + i], cc[i]);
  }
  __syncthreads();
  if (tid < 65) ws[OFF_P1 + blockIdx.x * 65 + tid] = acc[tid];
}

// Finalize BN1: exact mean/var from covariance (b1 cancels under BN).
__global__ void k_fin1(const float* __restrict__ W1, const float* __restrict__ g1,
                       const float* __restrict__ be1, const int* __restrict__ perms,
                       float* __restrict__ ws) {
  __shared__ float stats[65];
  __shared__ float xm[10];
  __shared__ float cov[10][10];
  int tid = threadIdx.x;
  if (tid < 65) {
    float t = 0.f;
    for (int b = 0; b < NB1; ++b) t += ws[OFF_P1 + b * 65 + tid];
    stats[tid] = t;
  }
  __syncthreads();
  if (tid < 10) xm[tid] = stats[tid] / (float)B_TOTAL;
  __syncthreads();
  if (tid < 100) {
    int i = tid / 10, j = tid % 10;
    int a = i < j ? i : j, b2 = i < j ? j : i;
    int idx = 10 * a - a * (a - 1) / 2 + (b2 - a);
    cov[i][j] = stats[10 + idx] / (float)B_TOTAL - xm[i] * xm[j];
  }
  __syncthreads();
  if (tid < 480) {
    int row = tid, p = row >> 2, c = row & 3;
    float w[10];
    #pragma unroll
    for (int k = 0; k < 10; ++k) {
      int q = k >> 1, d = k & 1;
      int j = 0;
      for (int jj = 0; jj < 5; ++jj) if (perms[p * 5 + jj] == q) j = jj;
      w[k] = W1[(2 * j + d) * 4 + c];
    }
    float mean = 0.f;
    #pragma unroll
    for (int k = 0; k < 10; ++k) mean += xm[k] * w[k];
    float var = 0.f;
    for (int i = 0; i < 10; ++i) {
      float t = 0.f;
      #pragma unroll
      for (int j = 0; j < 10; ++j) t += cov[i][j] * w[j];
      var += w[i] * t;
    }
    float sc = g1[c] * rsqrtf(var + EPSV);
    ws[OFF_S1 + row] = sc;
    ws[OFF_T1 + row] = be1[c] - mean * sc;
  }
}

// ---------------------------------------------------------------------------
// Phase 2: WMMA layer-1 GEMM (transposed), BN1+ReLU, layer 2, accumulate
// BN2 stats. mt (16 perm-channel rows) fixed per block => stats in registers.
__global__ void __launch_bounds__(256) k_phase2(const float* __restrict__ x,
                                                const float* __restrict__ W2,
                                                float* __restrict__ ws) {
  const int mt = blockIdx.x / C2;
  const int chunk = blockIdx.x % C2;
  const int tid = threadIdx.x;
  const int wave = tid >> 5, lane = tid & 31;
  const int hi = lane >> 4, bl = lane & 15;

  v2f A0 = *(const v2f*)(ws + OFF_W1T + ((mt * 3 + 0) * 32 + lane) * 2);
  v2f A1 = *(const v2f*)(ws + OFF_W1T + ((mt * 3 + 1) * 32 + lane) * 2);
  v2f A2 = *(const v2f*)(ws + OFF_W1T + ((mt * 3 + 2) * 32 + lane) * 2);

  const int rbase = mt * 16 + hi * 8;
  v4f s1a = *(const v4f*)(ws + OFF_S1 + rbase);
  v4f s1b = *(const v4f*)(ws + OFF_S1 + rbase + 4);
  v4f t1a = *(const v4f*)(ws + OFF_T1 + rbase);
  v4f t1b = *(const v4f*)(ws + OFF_T1 + rbase + 4);
  float s1v[8] = {s1a.x, s1a.y, s1a.z, s1a.w, s1b.x, s1b.y, s1b.z, s1b.w};
  float t1v[8] = {t1a.x, t1a.y, t1a.z, t1a.w, t1b.x, t1b.y, t1b.z, t1b.w};

  float w2[16];
  #pragma unroll
  for (int i = 0; i < 16; ++i) w2[i] = W2[i];

  float sum2[8], sq2[8];
  #pragma unroll
  for (int i = 0; i < 8; ++i) { sum2[i] = 0.f; sq2[i] = 0.f; }

  const int wgl = chunk * 8 + wave;     // 0..511
  const int NT = B_TOTAL / 16;          // 16384 tiles
  for (int tile = wgl; tile < NT; tile += C2 * 8) {
    const float* xb = x + (tile * 16 + bl) * 10;
    v2f B0 = *(const v2f*)(xb + 2 * hi);
    v2f B1 = *(const v2f*)(xb + 4 + 2 * hi);
    v2f B2;
    if (hi == 0) { B2 = *(const v2f*)(xb + 8); } else { B2.x = 0.f; B2.y = 0.f; }
    v8f D = {0.f, 0.f, 0.f, 0.f, 0.f, 0.f, 0.f, 0.f};
    D = __builtin_amdgcn_wmma_f32_16x16x4_f32(false, A0, false, B0, (short)0, D, false, false);
    D = __builtin_amdgcn_wmma_f32_16x16x4_f32(false, A1, false, B1, (short)0, D, false, false);
    D = __builtin_amdgcn_wmma_f32_16x16x4_f32(false, A2, false, B2, (short)0, D, false, false);
    float a1[8];
    #pragma unroll
    for (int i = 0; i < 8; ++i) a1[i] = fmaxf(0.f, D[i] * s1v[i] + t1v[i]);
    #pragma unroll
    for (int lp = 0; lp < 2; ++lp)
      #pragma unroll
      for (int cp = 0; cp < 4; ++cp) {
        float h = 0.f;
        #pragma unroll
        for (int c = 0; c < 4; ++c) h += a1[lp * 4 + c] * w2[c * 4 + cp];
        int ch = lp * 4 + cp;
        sum2[ch] += h; sq2[ch] += h * h;
      }
  }
  #pragma unroll
  for (int m = 1; m < 16; m <<= 1)
    #pragma unroll
    for (int i = 0; i < 8; ++i) {
      sum2[i] += __shfl_xor(sum2[i], m, 32);
      sq2[i]  += __shfl_xor(sq2[i],  m, 32);
    }
  __shared__ float Ls[16][8], Lq[16][8];
  if (bl == 0) {
    #pragma unroll
    for (int i = 0; i < 8; ++i) { Ls[hi * 8 + i][wave] = sum2[i]; Lq[hi * 8 + i][wave] = sq2[i]; }
  }
  __syncthreads();
  if (tid < 16) {
    float s = 0.f, q = 0.f;
    #pragma unroll
    for (int w = 0; w < 8; ++w) { s += Ls[tid][w]; q += Lq[tid][w]; }
    float* dst = ws + OFF_P2 + (mt * C2 + chunk) * 32;
    dst[tid] = s; dst[16 + tid] = q;
  }
}

__global__ void k_fin2(const float* __restrict__ g2, const float* __restrict__ be2,
                       float* __restrict__ ws) {
  int tid = threadIdx.x;
  if (tid < 480) {
    int mt = tid / 16, chl = tid % 16;
    float s = 0.f, q = 0.f;
    for (int ch = 0; ch < C2; ++ch) {
      const float* src = ws + OFF_P2 + (mt * C2 + ch) * 32;
      s += src[chl]; q += src[16 + chl];
    }
    float mean = s / (float)B_TOTAL;
    float var = q / (float)B_TOTAL - mean * mean;
    int c = tid & 3;
    float sc = g2[c] * rsqrtf(var + EPSV);
    ws[OFF_S2 + tid] = sc;
    ws[OFF_T2 + tid] = be2[c] - mean * sc;
  }
}

// ---------------------------------------------------------------------------
// Phase 3: full forward. One wave per 16-batch tile, loop over all 30 mt
// groups (all 120 perms), parity-signed accumulation in-lane, no atomics.
__global__ void __launch_bounds__(256) k_phase3(const float* __restrict__ x,
                                                const float* __restrict__ W2,
                                                const float* __restrict__ W3,
                                                const float* __restrict__ b3,
                                                const int* __restrict__ parity,
                                                const float* __restrict__ ws,
                                                float* __restrict__ out) {
  const int tid = threadIdx.x;
  const int wave = tid >> 5, lane = tid & 31;
  const int hi = lane >> 4, bl = lane & 15;
  const int tile = blockIdx.x * 8 + wave;
  const float* xb = x + (tile * 16 + bl) * 10;
  v2f B0 = *(const v2f*)(xb + 2 * hi);
  v2f B1 = *(const v2f*)(xb + 4 + 2 * hi);
  v2f B2;
  if (hi == 0) { B2 = *(const v2f*)(xb + 8); } else { B2.x = 0.f; B2.y = 0.f; }

  float w2[16];
  #pragma unroll
  for (int i = 0; i < 16; ++i) w2[i] = W2[i];
  float w3[4];
  #pragma unroll
  for (int i = 0; i < 4; ++i) w3[i] = W3[i];
  const float bb3 = b3[0];

  float accs = 0.f;
  for (int mt = 0; mt < 30; ++mt) {
    v2f A0 = *(const v2f*)(ws + OFF_W1T + ((mt * 3 + 0) * 32 + lane) * 2);
    v2f A1 = *(const v2f*)(ws + OFF_W1T + ((mt * 3 + 1) * 32 + lane) * 2);
    v2f A2 = *(const v2f*)(ws + OFF_W1T + ((mt * 3 + 2) * 32 + lane) * 2);
    v8f D = {0.f, 0.f, 0.f, 0.f, 0.f, 0.f, 0.f, 0.f};
    D = __builtin_amdgcn_wmma_f32_16x16x4_f32(false, A0, false, B0, (short)0, D, false, false);
    D = __builtin_amdgcn_wmma_f32_16x16x4_f32(false, A1, false, B1, (short)0, D, false, false);
    D = __builtin_amdgcn_wmma_f32_16x16x4_f32(false, A2, false, B2, (short)0, D, false, false);

    const int rbase = mt * 16 + hi * 8;
    v4f s1a = *(const v4f*)(ws + OFF_S1 + rbase);
    v4f s1b = *(const v4f*)(ws + OFF_S1 + rbase + 4);
    v4f t1a = *(const v4f*)(ws + OFF_T1 + rbase);
    v4f t1b = *(const v4f*)(ws + OFF_T1 + rbase + 4);
    v4f s2a = *(const v4f*)(ws + OFF_S2 + rbase);
    v4f s2b = *(const v4f*)(ws + OFF_S2 + rbase + 4);
    v4f t2a = *(const v4f*)(ws + OFF_T2 + rbase);
    v4f t2b = *(const v4f*)(ws + OFF_T2 + rbase + 4);
    float s1v[8] = {s1a.x, s1a.y, s1a.z, s1a.w, s1b.x, s1b.y, s1b.z, s1b.w};
    float t1v[8] = {t1a.x, t1a.y, t1a.z, t1a.w, t1b.x, t1b.y, t1b.z, t1b.w};
    float s2v[8] = {s2a.x, s2a.y, s2a.z, s2a.w, s2b.x, s2b.y, s2b.z, s2b.w};
    float t2v[8] = {t2a.x, t2a.y, t2a.z, t2a.w, t2b.x, t2b.y, t2b.z, t2b.w};

    float a1[8];
    #pragma unroll
    for (int i = 0; i < 8; ++i) a1[i] = fmaxf(0.f, D[i] * s1v[i] + t1v[i]);

    #pragma unroll
    for (int lp = 0; lp < 2; ++lp) {
      float u = bb3;
      #pragma unroll
      for (int cp = 0; cp < 4; ++cp) {
        float h = 0.f;
        #pragma unroll
        for (int c = 0; c < 4; ++c) h += a1[lp * 4 + c] * w2[c * 4 + cp];
        float z = fmaxf(0.f, h * s2v[lp * 4 + cp] + t2v[lp * 4 + cp]);
        u += z * w3[cp];
      }
      float y = sqrtf(fabsf(tanhf(u)));
      int p = mt * 4 + hi * 2 + lp;
      accs += (float)parity[p] * y;
    }
  }
  accs += __shfl_xor(accs, 16, 32);
  if (hi == 0) out[tile * 16 + bl] = accs / 120.f;
}

// ---------------------------------------------------------------------------
extern "C" void kernel_launch(void* const* d_in, const int* in_sizes, int n_in,
                              void* d_out, int out_size, void* d_ws, size_t ws_size,
                              hipStream_t stream) {
  (void)in_sizes; (void)n_in; (void)out_size; (void)ws_size;
  const float* x    = (const float*)d_in[0];
  const float* W1   = (const float*)d_in[1];
  // d_in[2] = b1, d_in[6] = b2: cancel exactly under BatchNorm, unused.
  const float* g1   = (const float*)d_in[3];
  const float* be1  = (const float*)d_in[4];
  const float* W2   = (const float*)d_in[5];
  const float* g2   = (const float*)d_in[7];
  const float* be2  = (const float*)d_in[8];
  const float* W3   = (const float*)d_in[9];
  const float* b3   = (const float*)d_in[10];
  const int* perms  = (const int*)d_in[11];
  const int* parity = (const int*)d_in[12];
  float* ws  = (float*)d_ws;
  float* out = (float*)d_out;

  k_setup_w1t<<<12, 256, 0, stream>>>(W1, perms, ws);
  k_phase1<<<NB1, 256, 0, stream>>>(x, ws);
  k_fin1<<<1, 512, 0, stream>>>(W1, g1, be1, perms, ws);
  k_phase2<<<30 * C2, 256, 0, stream>>>(x, W2, ws);
  k_fin2<<<1, 512, 0, stream>>>(g2, be2, ws);
  k_phase3<<<(B_TOTAL / 16) / 8, 256, 0, stream>>>(x, W2, W3, b3, parity, ws, out);
}